// LightGCN_369367188130
// MI455X (gfx1250) — compile-verified
//
#include <hip/hip_runtime.h>

#define EMB 128

typedef __attribute__((ext_vector_type(2))) float v2f;
typedef __attribute__((ext_vector_type(8))) float v8f;

// ---- init: rep_a = embedding, acc = embedding (float4 vectorized) ----
__global__ void lgcn_init_kernel(const float4* __restrict__ src,
                                 float4* __restrict__ rep,
                                 float4* __restrict__ acc, int n4) {
    int i = blockIdx.x * blockDim.x + threadIdx.x;
    if (i < n4) { float4 v = src[i]; rep[i] = v; acc[i] = v; }
}

// ---- zero the scatter target ----
__global__ void lgcn_zero_kernel(float4* __restrict__ p, int n4) {
    int i = blockIdx.x * blockDim.x + threadIdx.x;
    if (i < n4) { p[i] = make_float4(0.f, 0.f, 0.f, 0.f); }
}

// ---- scatter: one edge per wave32; 4 f32 atomics per lane cover 128 dims ----
__global__ void lgcn_scatter_kernel(const float* __restrict__ rep_src,
                                    const int* __restrict__ esrc,
                                    const int* __restrict__ edst,
                                    const float* __restrict__ ew,
                                    float* __restrict__ rep_dst, int n_edges) {
    int e    = blockIdx.x * 8 + (threadIdx.x >> 5);   // 8 wave32 per 256-thread block
    int lane = threadIdx.x & 31;
    if (e >= n_edges) return;
    int   s = esrc[e];
    int   d = edst[e];
    float w = ew[e];
    const float4* srow = (const float4*)(rep_src + (size_t)s * EMB);
    float4 v = srow[lane];                            // 32 lanes x float4 = 128 floats
    float* dp = rep_dst + (size_t)d * EMB + lane * 4;
    atomicAdd(dp + 0, v.x * w);
    atomicAdd(dp + 1, v.y * w);
    atomicAdd(dp + 2, v.z * w);
    atomicAdd(dp + 3, v.w * w);
}

// ---- acc += rep ----
__global__ void lgcn_accadd_kernel(float4* __restrict__ acc,
                                   const float4* __restrict__ rep, int n4) {
    int i = blockIdx.x * blockDim.x + threadIdx.x;
    if (i < n4) {
        float4 a = acc[i]; float4 r = rep[i];
        a.x += r.x; a.y += r.y; a.z += r.z; a.w += r.w;
        acc[i] = a;
    }
}

// ---- gather user rows (float4 vectorized: 32 threads per user) ----
__global__ void lgcn_gather_kernel(const float4* __restrict__ acc,
                                   const int* __restrict__ users,
                                   float4* __restrict__ users_r, int nq) {
    int t = blockIdx.x * blockDim.x + threadIdx.x;
    int u = t >> 5, l = t & 31;
    if (u < nq) users_r[(size_t)u * 32 + l] = acc[(size_t)users[u] * 32 + l];
}

// ---- GEMM: scores = (U x I^T) / 16 via V_WMMA_F32_16X16X4_F32 ----
// One wave32 computes a 64x16 strip: 4 M-tiles sharing one B operand per
// K-step (5 loads / 4 WMMAs -> 6.4 FLOP per L2 byte). K=128 -> 128 WMMAs.
// A 16x4 layout: lane l holds A[l%16][k0 + 2*(l/16) + {0,1}]
// B 4x16 layout: lane l holds B[k0 + 2*(l/16) + {0,1}][l%16] = I[l%16][same k]
// C/D 16x16:     lane l, vgpr j holds D[j + 8*(l/16)][l%16]
__global__ void lgcn_gemm_kernel(const float* __restrict__ U,
                                 const float* __restrict__ I,
                                 float* __restrict__ out,
                                 int m_strips, int n_items, int total_waves) {
    int wave = (blockIdx.x * blockDim.x + threadIdx.x) >> 5;
    if (wave >= total_waves) return;                  // wave-uniform: EXEC stays all-ones
    int lane = threadIdx.x & 31;
    int ms = wave % m_strips;                         // 64-row strip of U
    int nt = wave / m_strips;                         // 16-col tile of I
    int half = lane >> 4;                             // 0: k+{0,1}, 1: k+{2,3}
    int r    = lane & 15;

    const float* up = U + (size_t)(ms * 64 + r) * EMB + 2 * half;
    const float* ip = I + (size_t)(nt * 16 + r) * EMB + 2 * half;

    v8f c0 = {}, c1 = {}, c2 = {}, c3 = {};
#pragma unroll
    for (int k = 0; k < EMB; k += 4) {
        v2f b  = *(const v2f*)(ip + k);               // shared across 4 WMMAs
        v2f a0 = *(const v2f*)(up + k);
        v2f a1 = *(const v2f*)(up + 16 * EMB + k);
        v2f a2 = *(const v2f*)(up + 32 * EMB + k);
        v2f a3 = *(const v2f*)(up + 48 * EMB + k);
        c0 = __builtin_amdgcn_wmma_f32_16x16x4_f32(false, a0, false, b, (short)0, c0, false, false);
        c1 = __builtin_amdgcn_wmma_f32_16x16x4_f32(false, a1, false, b, (short)0, c1, false, false);
        c2 = __builtin_amdgcn_wmma_f32_16x16x4_f32(false, a2, false, b, (short)0, c2, false, false);
        c3 = __builtin_amdgcn_wmma_f32_16x16x4_f32(false, a3, false, b, (short)0, c3, false, false);
    }

    int n = nt * 16 + r;
    float* orow = out + (size_t)(ms * 64 + half * 8) * n_items + n;
#pragma unroll
    for (int j = 0; j < 8; ++j) {
        orow[(size_t)(j +  0) * n_items] = c0[j] * 0.0625f;   // fold (1/4)*(1/4)
        orow[(size_t)(j + 16) * n_items] = c1[j] * 0.0625f;
        orow[(size_t)(j + 32) * n_items] = c2[j] * 0.0625f;
        orow[(size_t)(j + 48) * n_items] = c3[j] * 0.0625f;
    }
}

extern "C" void kernel_launch(void* const* d_in, const int* in_sizes, int n_in,
                              void* d_out, int out_size, void* d_ws, size_t ws_size,
                              hipStream_t stream) {
    const float* embedding = (const float*)d_in[0];
    const float* edge_w    = (const float*)d_in[1];
    const int*   edge_src  = (const int*)d_in[2];
    const int*   edge_dst  = (const int*)d_in[3];
    const int*   users     = (const int*)d_in[4];
    float*       out       = (float*)d_out;

    // All shape parameters derived host-side from sizes (no device scalar read).
    const int n_nodes     = in_sizes[0] / EMB;     // 100000
    const int n_edges     = in_sizes[1];           // 3200000
    const int n_q         = in_sizes[4];           // 1024
    const int n_items     = out_size / n_q;        // 50000
    const int n_users_val = n_nodes - n_items;     // 50000

    const size_t NE = (size_t)n_nodes * EMB;       // 12.8M floats per buffer
    float* rep_a   = (float*)d_ws;
    float* rep_b   = rep_a + NE;
    float* accb    = rep_b + NE;
    float* users_r = accb + NE;

    const int n4      = (int)(NE / 4);
    const int blocks4 = (n4 + 255) / 256;

    lgcn_init_kernel<<<blocks4, 256, 0, stream>>>((const float4*)embedding,
                                                  (float4*)rep_a, (float4*)accb, n4);

    float* cur = rep_a;
    float* nxt = rep_b;
    for (int layer = 0; layer < 3; ++layer) {
        lgcn_zero_kernel<<<blocks4, 256, 0, stream>>>((float4*)nxt, n4);
        lgcn_scatter_kernel<<<(n_edges + 7) / 8, 256, 0, stream>>>(
            cur, edge_src, edge_dst, edge_w, nxt, n_edges);
        lgcn_accadd_kernel<<<blocks4, 256, 0, stream>>>((float4*)accb,
                                                        (const float4*)nxt, n4);
        float* t = cur; cur = nxt; nxt = t;
    }

    lgcn_gather_kernel<<<(n_q * 32 + 255) / 256, 256, 0, stream>>>(
        (const float4*)accb, users, (float4*)users_r, n_q);

    const int m_strips    = n_q / 64;              // 16 strips of 64 rows
    const int n_tiles     = n_items / 16;          // 3125
    const int total_waves = m_strips * n_tiles;    // 50000 (divisible by 8)
    lgcn_gemm_kernel<<<(total_waves + 7) / 8, 256, 0, stream>>>(
        users_r, accb + (size_t)n_users_val * EMB, out, m_strips, n_items, total_waves);
}